// LightweightConv1dTBC_34059090657410
// MI455X (gfx1250) — compile-verified
//
#include <hip/hip_runtime.h>

typedef __attribute__((ext_vector_type(2))) float v2f;
typedef __attribute__((ext_vector_type(8))) float v8f;

#define T_DIM 2048
#define B_DIM 8
#define C_DIM 1024
#define K_DIM 31
#define P_PAD 15
#define R_DIM 64                    /* channels per head */
#define STRIDE (B_DIM * C_DIM)      /* floats per timestep = 8192 */
#define TPW 8                       /* tiles (of 16 timesteps) per wave */
#define N_TILES (T_DIM / 16)        /* 128 */

__device__ __forceinline__ void ld_stile(v2f d[4], const float* base) {
#pragma unroll
  for (int ks = 0; ks < 4; ++ks) {          // imm-offset b32 loads
    d[ks].x = base[(size_t)(ks * 4)     * STRIDE];
    d[ks].y = base[(size_t)(ks * 4 + 1) * STRIDE];
  }
}

__device__ __forceinline__ void zero_stile(v2f d[4]) {
#pragma unroll
  for (int ks = 0; ks < 4; ++ks) { d[ks].x = 0.0f; d[ks].y = 0.0f; }
}

// One 8-tile chunk along T. EDGE_LO: first chunk (leading s-tile is zeros).
// EDGE_HI: last chunk (refills past T are zeros, at compile-time unroll idx).
// Interior waves (14/16): branch-free, mask-free streaming.
template <bool EDGE_LO, bool EDGE_HI>
__device__ __forceinline__ void run_chunk(const float* __restrict__ xchunk,
                                          float* __restrict__ ochunk,
                                          const v2f afr[12], float bv) {
  v2f bwin[3][4];   // rolling window of 3 s-tiles

  if (EDGE_LO) zero_stile(bwin[0]);
  else         ld_stile(bwin[0], xchunk - (size_t)16 * STRIDE);
  ld_stile(bwin[1], xchunk);
  ld_stile(bwin[2], xchunk + (size_t)16 * STRIDE);

#pragma unroll
  for (int it = 0; it < TPW; ++it) {
    // Refill for two tiles ahead, issued BEFORE the WMMA chain (SSA renames
    // the slot, so loads overlap this tile's WMMAs).
    v2f nxt[4];
    if (it < TPW - 1) {
      if (EDGE_HI && (2 + it) >= TPW) zero_stile(nxt);      // compile-time
      else ld_stile(nxt, xchunk + (size_t)(2 + it) * 16 * STRIDE);
    }

    // C starts as inline 0; bias is added at store time.
    v8f acc = {};
#pragma unroll
    for (int st = 0; st < 3; ++st) {
      v2f* bt = bwin[(it + st) % 3];                         // consts after unroll
#pragma unroll
      for (int ks = 0; ks < 4; ++ks) {
        acc = __builtin_amdgcn_wmma_f32_16x16x4_f32(
            /*neg_a=*/false, afr[st * 4 + ks], /*neg_b=*/false, bt[ks],
            /*c_mod=*/(short)0, acc, /*reuse_a=*/false, /*reuse_b=*/false);
      }
    }

    // D layout: VGPR r = rows M=r (lanes 0-15) / M=r+8 (lanes 16-31).
    float* po = ochunk + (size_t)it * 16 * STRIDE;
#pragma unroll
    for (int r = 0; r < 8; ++r)
      __builtin_nontemporal_store(acc[r] + bv, po + (size_t)r * STRIDE);

    if (it < TPW - 1) {
#pragma unroll
      for (int ks = 0; ks < 4; ++ks) bwin[it % 3][ks] = nxt[ks];
    }
  }
}

// Block: 32x8 = 8 waves; wave y owns an 8-tile chunk along T at fixed
// (batch, 16-channel tile). out_tile = Toeplitz(w)[16x48] * X[48x16] via
// 12x V_WMMA_F32_16X16X4_F32 per tile; rolling window -> x read exactly once.
__global__ __launch_bounds__(256) void
LightweightConv1dTBC_kernel(const float* __restrict__ x,
                            const float* __restrict__ weight,
                            const float* __restrict__ bias,
                            float* __restrict__ out) {
  __shared__ float wsm[K_DIM];

  const int b    = blockIdx.y;
  const int c0   = blockIdx.z * 16;
  const int head = c0 / R_DIM;

  // Softmax of weight[head,0,:] (wave-uniform scalar loads; redundant compute).
  {
    const float* wr = weight + head * K_DIM;
    float mx = wr[0];
#pragma unroll
    for (int k = 1; k < K_DIM; ++k) mx = fmaxf(mx, wr[k]);
    float sum = 0.0f;
#pragma unroll
    for (int k = 0; k < K_DIM; ++k) sum += __expf(wr[k] - mx);
    const float inv = 1.0f / sum;
    const int tid = threadIdx.y * 32 + threadIdx.x;
    if (tid < K_DIM) wsm[tid] = __expf(wr[tid] - mx) * inv;
  }
  __syncthreads();   // EXEC all-ones below (WMMA requirement)

  const int lane = threadIdx.x;
  const int mN   = lane & 15;      // M for A, N (channel) for B/D
  const int hi   = lane >> 4;      // half-wave select

  // ---- A fragments: banded Toeplitz of the 31 taps; t-invariant, built once.
  v2f afr[12];
#pragma unroll
  for (int st = 0; st < 3; ++st) {
#pragma unroll
    for (int ks = 0; ks < 4; ++ks) {
      const int i  = st * 4 + ks;
      const int j0 = ks * 4 + hi * 2;
      const int d0 = (st - 1) * 16 + j0 - mN + P_PAD;
      const int d1 = d0 + 1;
      const float w0 = wsm[min(max(d0, 0), K_DIM - 1)];
      const float w1 = wsm[min(max(d1, 0), K_DIM - 1)];
      afr[i].x = ((unsigned)d0 < (unsigned)K_DIM) ? w0 : 0.0f;
      afr[i].y = ((unsigned)d1 < (unsigned)K_DIM) ? w1 : 0.0f;
    }
  }
  const float bv = bias[c0 + mN];

  __builtin_amdgcn_sched_barrier(0);   // pin softmax/A-frag/ds work in prologue

  // ---- Per-lane chunk base pointers (everything else is imm offsets) ----
  const int tb0 = (blockIdx.x * 8 + threadIdx.y) * TPW;   // first tile index
  const size_t colBase = (size_t)b * C_DIM + c0 + mN;
  const float* xchunk =
      x + colBase + (size_t)(hi * 2) * STRIDE + (size_t)tb0 * 16 * STRIDE;
  float* ochunk =
      out + colBase + (size_t)(hi * 8) * STRIDE + (size_t)tb0 * 16 * STRIDE;

  // Scalar 3-way dispatch (readfirstlane -> SGPR -> s_cbranch, EXEC untouched).
  const int tb0u = __builtin_amdgcn_readfirstlane(tb0);
  if (tb0u == 0)                      run_chunk<true,  false>(xchunk, ochunk, afr, bv);
  else if (tb0u == N_TILES - TPW)     run_chunk<false, true >(xchunk, ochunk, afr, bv);
  else                                run_chunk<false, false>(xchunk, ochunk, afr, bv);
}

extern "C" void kernel_launch(void* const* d_in, const int* in_sizes, int n_in,
                              void* d_out, int out_size, void* d_ws, size_t ws_size,
                              hipStream_t stream) {
  (void)in_sizes; (void)n_in; (void)d_ws; (void)ws_size; (void)out_size;
  const float* x      = (const float*)d_in[0];   // (T, B, C) f32
  const float* weight = (const float*)d_in[1];   // (H, 1, K) f32
  const float* bias   = (const float*)d_in[2];   // (C,) f32
  float* out          = (float*)d_out;           // (T, B, C) f32

  dim3 block(32, 8);                                  // 8 waves (wave32)
  dim3 grid(N_TILES / (8 * TPW), B_DIM, C_DIM / 16);  // (2, 8, 64)
  LightweightConv1dTBC_kernel<<<grid, block, 0, stream>>>(x, weight, bias, out);
}